// PartialFC_V2_57002805953021
// MI455X (gfx1250) — compile-verified
//
#include <hip/hip_runtime.h>
#include <math.h>

typedef __attribute__((ext_vector_type(2))) float v2f;
typedef __attribute__((ext_vector_type(8))) float v8f;

#define EMB      512
#define BATCH    512
#define NCLS     100000
#define TILE_M   64
#define TILE_N   128
#define KT       32
#define KTP      36                      // padded LDS row stride (floats): 16B aligned, bank-friendly
#define NT_TILES ((NCLS + TILE_N - 1) / TILE_N)   // 782

// dynamic-LDS layout (floats). Stage buffers are double-buffered and aliased by Lg.
#define SA_FLTS   (TILE_M * KTP)                 // 2304 per buffer
#define SW_FLTS   (TILE_N * KTP)                 // 4608 per buffer
#define SW_BASE   (2 * SA_FLTS)                  // 4608
#define RED_BASE  (SW_BASE + 2 * SW_FLTS)        // 13824
#define LGS       (TILE_N + 2)                   // 130: Lg row stride, Lg aliases [0, 8320)
#define SMEM_FLTS (RED_BASE + 2 * 4 * TILE_M)    // 14336 floats = 57344 bytes

// ---------------- embedding row L2-normalize (margin branch only) ----------------
__global__ void pfc_norm_emb(const float* __restrict__ emb, float* __restrict__ normA,
                             const int* __restrict__ epoch) {
  if (*epoch <= 30) return;
  __shared__ float sh[256];
  const int r = blockIdx.x, t = threadIdx.x;
  float x0 = emb[r * EMB + t];
  float x1 = emb[r * EMB + t + 256];
  sh[t] = x0 * x0 + x1 * x1;
  __syncthreads();
  for (int s = 128; s > 0; s >>= 1) { if (t < s) sh[t] += sh[t + s]; __syncthreads(); }
  const float scale = 1.0f / fmaxf(sqrtf(sh[0]), 1e-12f);
  normA[r * EMB + t]       = x0 * scale;
  normA[r * EMB + t + 256] = x1 * scale;
}

// ---------------- per-class 1/||w|| (one wave32 per class) ----------------
__global__ void pfc_wnorm(const float* __restrict__ w, float* __restrict__ invw,
                          const int* __restrict__ epoch) {
  if (*epoch <= 30) return;
  const int wave = threadIdx.x >> 5, lane = threadIdx.x & 31;
  const int c = blockIdx.x * 8 + wave;
  const float* row = w + (size_t)c * EMB;
  float s = 0.f;
  for (int i = lane * 4; i < EMB; i += 128) {
    float a = row[i], b = row[i + 1], d = row[i + 2], e = row[i + 3];
    s += a * a + b * b + d * d + e * e;
  }
  for (int off = 16; off > 0; off >>= 1) s += __shfl_xor(s, off, 32);
  if (lane == 0) invw[c] = 1.0f / fmaxf(sqrtf(s), 1e-12f);
}

// ---------------- tiled WMMA GEMM + streaming (flash) softmax partials ----------------
// Block: 256 threads = 8 waves. Tile: 64 rows x 128 classes, full K=512.
// Double-buffered LDS stages filled by async global->LDS DMA (GVS form: scalar base + small
// chunk-invariant per-thread voffset); WMMAs overlap prefetch; frag loads pipelined 1 k-step.
__global__ __launch_bounds__(256)
void pfc_gemm_softmax(const float* __restrict__ A, const float* __restrict__ W,
                      const float* __restrict__ invw, const int* __restrict__ labels,
                      float* __restrict__ pmax, float* __restrict__ psum,
                      float* __restrict__ ztarg, const int* __restrict__ epoch,
                      int want_margin) {
  const int e = *epoch;
  if (want_margin ? (e <= 30) : (e > 30)) return;

  extern __shared__ float smem[];

  const int tid   = threadIdx.x;
  const int lane  = tid & 31;
  const int wave  = tid >> 5;
  const int m0    = blockIdx.y * TILE_M;
  const int n0    = blockIdx.x * TILE_N;
  const int m0w   = (wave & 3) * 16;     // wave row offset in tile
  const int n0w   = (wave >> 2) * 64;    // wave col offset in tile
  const int nl    = lane & 15;
  const int khalf = (lane >> 4) * 2;     // A/B frag K split across lane halves

  // ---- chunk-invariant per-thread staging state (voffsets in bytes, LDS offsets for buf0) ----
  int      voffA[2], voffW[4];
  unsigned ldsA[2], ldsW[4];
  bool     okW[4];
#pragma unroll
  for (int it = 0; it < 2; ++it) {
    int idx = tid + it * 256;
    int row = idx >> 3, kq = idx & 7;
    voffA[it] = ((m0 + row) * EMB + kq * 4) * 4;
    ldsA[it]  = (unsigned)(unsigned long long)(&smem[row * KTP + kq * 4]);
  }
#pragma unroll
  for (int it = 0; it < 4; ++it) {
    int idx = tid + it * 256;
    int row = idx >> 3, kq = idx & 7;
    voffW[it] = (row * EMB + kq * 4) * 4;
    ldsW[it]  = (unsigned)(unsigned long long)(&smem[SW_BASE + row * KTP + kq * 4]);
    okW[it]   = (n0 + row) < NCLS;
  }
  const unsigned long long baseA0 = (unsigned long long)A;                       // + ks*4 per chunk
  const unsigned long long baseW0 = (unsigned long long)(W + (size_t)n0 * EMB);  // + ks*4 per chunk

  // issue async global->LDS copies for one K-chunk into stage buffer `buf`
  auto stage = [&](int buf, int ks) {
    const unsigned long long bA = baseA0 + (unsigned long long)(ks * 4);
    const unsigned long long bW = baseW0 + (unsigned long long)(ks * 4);
    const unsigned abump = buf * (SA_FLTS * 4);
    const unsigned wbump = buf * (SW_FLTS * 4);
#pragma unroll
    for (int it = 0; it < 2; ++it) {
      asm volatile("global_load_async_to_lds_b128 %0, %1, %2"
                   :: "v"(ldsA[it] + abump), "v"(voffA[it]), "s"(bA) : "memory");
    }
#pragma unroll
    for (int it = 0; it < 4; ++it) {
      if (okW[it]) {
        asm volatile("global_load_async_to_lds_b128 %0, %1, %2"
                     :: "v"(ldsW[it] + wbump), "v"(voffW[it]), "s"(bW) : "memory");
      } else {
        // zero-pad classes past NCLS (plain LDS store, waited by the chunk barrier)
        float* lp = smem + SW_BASE + buf * SW_FLTS;
        int idx = tid + it * 256;
        int row = idx >> 3, kq = idx & 7;
        *(float4*)&lp[row * KTP + kq * 4] = make_float4(0.f, 0.f, 0.f, 0.f);
      }
    }
  };

  v8f acc[4] = {v8f{}, v8f{}, v8f{}, v8f{}};

  stage(0, 0);
  asm volatile("s_wait_asynccnt 0" ::: "memory");
  __syncthreads();

  const int nchunks = EMB / KT;   // 16
  for (int ch = 0; ch < nchunks; ++ch) {
    const int buf = ch & 1;
    if (ch + 1 < nchunks) stage(buf ^ 1, (ch + 1) * KT);   // prefetch next chunk (async)

    const float* As = smem + buf * SA_FLTS;
    const float* Ws = smem + SW_BASE + buf * SW_FLTS;
    const float* ap = &As[(m0w + nl) * KTP + khalf];
    const float* bp = &Ws[(n0w + nl) * KTP + khalf];

    // software-pipelined fragment reads: load k-step kk+4 while WMMAing kk
    v2f a  = *(const v2f*)(ap);
    v2f b0 = *(const v2f*)(bp + 0  * KTP);
    v2f b1 = *(const v2f*)(bp + 16 * KTP);
    v2f b2 = *(const v2f*)(bp + 32 * KTP);
    v2f b3 = *(const v2f*)(bp + 48 * KTP);
#pragma unroll
    for (int kk = 0; kk < KT; kk += 4) {
      v2f an = a, b0n = b0, b1n = b1, b2n = b2, b3n = b3;
      if (kk + 4 < KT) {
        an  = *(const v2f*)(ap + kk + 4);
        b0n = *(const v2f*)(bp + 0  * KTP + kk + 4);
        b1n = *(const v2f*)(bp + 16 * KTP + kk + 4);
        b2n = *(const v2f*)(bp + 32 * KTP + kk + 4);
        b3n = *(const v2f*)(bp + 48 * KTP + kk + 4);
      }
      acc[0] = __builtin_amdgcn_wmma_f32_16x16x4_f32(false, a, false, b0, (short)0, acc[0], false, false);
      acc[1] = __builtin_amdgcn_wmma_f32_16x16x4_f32(false, a, false, b1, (short)0, acc[1], false, false);
      acc[2] = __builtin_amdgcn_wmma_f32_16x16x4_f32(false, a, false, b2, (short)0, acc[2], false, false);
      acc[3] = __builtin_amdgcn_wmma_f32_16x16x4_f32(false, a, false, b3, (short)0, acc[3], false, false);
      a = an; b0 = b0n; b1 = b1n; b2 = b2n; b3 = b3n;
    }
    asm volatile("s_wait_asynccnt 0" ::: "memory");   // prefetch landed
    __syncthreads();                                  // all waves done reading `buf`
  }

  // ---- epilogue: normalize by 1/||w||, ArcFace margin on target, write tile to LDS ----
  // Lg (64 x 130 f32) aliases the stage buffers: safe, all stage reads retired at last barrier.
  float* Lg = smem;
  const float COSM = 0.87758256189037276f;   // cos(0.5)
  const float SINM = 0.47942553860420301f;   // sin(0.5)
  const float THR  = -0.87758256189037276f;  // cos(pi - 0.5)
  const float MMC  = 0.23971276930210151f;   // sin(pi - 0.5)*0.5
  const int rbase  = m0w + ((lane >> 4) << 3); // C/D: VGPR r -> row = r + 8*(lane/16)

#pragma unroll
  for (int f = 0; f < 4; ++f) {
    const int col  = n0w + f * 16 + nl;
    const int gcol = n0 + col;
    float iw = 1.0f;
    if (want_margin) iw = (gcol < NCLS) ? invw[gcol] : 0.0f;
#pragma unroll
    for (int r = 0; r < 8; ++r) {
      const int row  = rbase + r;
      const int grow = m0 + row;
      const int lab  = labels[grow];
      float z;
      if (want_margin) {
        float cosv = acc[f][r] * iw;
        if (gcol == lab) {
          float tt = cosv;
          float sn = sqrtf(fmaxf(1.0f - tt * tt, 0.0f));
          float wm = tt * COSM - sn * SINM;          // cos(theta + m)
          float nt = (tt > THR) ? wm : (tt - MMC);   // easy-margin safeguard
          z = 64.0f * nt;
          ztarg[grow] = z;                            // unique writer per row
        } else {
          z = 64.0f * cosv;
        }
      } else {
        z = acc[f][r];
        if (gcol == lab) ztarg[grow] = z;
      }
      if (gcol >= NCLS) z = -1e30f;                   // pad columns -> exp() == 0
      Lg[row * LGS + col] = z;
    }
  }
  __syncthreads();

  // ---- per-row partial (max, sum exp) over this 128-col tile ----
  float* redm = smem + RED_BASE;             // [4][64]
  float* reds = smem + RED_BASE + 4 * TILE_M;
  const int row  = tid & 63;
  const int part = tid >> 6;
  float mx = -1e30f;
#pragma unroll 8
  for (int c = part * 32; c < part * 32 + 32; ++c) mx = fmaxf(mx, Lg[row * LGS + c]);
  float sm = 0.f;
#pragma unroll 8
  for (int c = part * 32; c < part * 32 + 32; ++c) {
    float z = Lg[row * LGS + c];
    if (z > -1e29f) sm += expf(z - mx);
  }
  redm[part * TILE_M + row] = mx;
  reds[part * TILE_M + row] = sm;
  __syncthreads();
  if (tid < 64) {
    float M = fmaxf(fmaxf(redm[0 * TILE_M + tid], redm[1 * TILE_M + tid]),
                    fmaxf(redm[2 * TILE_M + tid], redm[3 * TILE_M + tid]));
    float S = 0.f;
#pragma unroll
    for (int p = 0; p < 4; ++p) {
      float sp = reds[p * TILE_M + tid];
      if (sp > 0.f) S += sp * expf(redm[p * TILE_M + tid] - M);
    }
    const size_t off = (size_t)(m0 + tid) * NT_TILES + blockIdx.x;
    pmax[off] = M;
    psum[off] = S;
  }
}

// ---------------- combine tile partials per row: loss_r = logsumexp - z_target ----------------
__global__ void pfc_row_reduce(const float* __restrict__ pmax, const float* __restrict__ psum,
                               const float* __restrict__ ztarg, float* __restrict__ rowloss) {
  __shared__ float sh[256];
  const int r = blockIdx.x, t = threadIdx.x;
  const float* pm = pmax + (size_t)r * NT_TILES;
  const float* ps = psum + (size_t)r * NT_TILES;
  float mx = -1e30f;
  for (int i = t; i < NT_TILES; i += 256) mx = fmaxf(mx, pm[i]);
  sh[t] = mx; __syncthreads();
  for (int s = 128; s > 0; s >>= 1) { if (t < s) sh[t] = fmaxf(sh[t], sh[t + s]); __syncthreads(); }
  const float M = sh[0]; __syncthreads();
  float ls = 0.f;
  for (int i = t; i < NT_TILES; i += 256) {
    float sp = ps[i];
    if (sp > 0.f) ls += sp * expf(pm[i] - M);
  }
  sh[t] = ls; __syncthreads();
  for (int s = 128; s > 0; s >>= 1) { if (t < s) sh[t] += sh[t + s]; __syncthreads(); }
  if (t == 0) rowloss[r] = M + logf(sh[0]) - ztarg[r];
}

// ---------------- mean over rows -> scalar loss ----------------
__global__ void pfc_final(const float* __restrict__ rowloss, float* __restrict__ out) {
  __shared__ float sh[256];
  const int t = threadIdx.x;
  sh[t] = rowloss[t] + rowloss[t + 256];
  __syncthreads();
  for (int s = 128; s > 0; s >>= 1) { if (t < s) sh[t] += sh[t + s]; __syncthreads(); }
  if (t == 0) out[0] = sh[0] / (float)BATCH;
}

extern "C" void kernel_launch(void* const* d_in, const int* in_sizes, int n_in,
                              void* d_out, int out_size, void* d_ws, size_t ws_size,
                              hipStream_t stream) {
  (void)in_sizes; (void)n_in; (void)out_size; (void)ws_size;
  const float* emb    = (const float*)d_in[0];   // [512, 512]
  const float* w      = (const float*)d_in[1];   // [100000, 512]
  const float* fcw    = (const float*)d_in[2];   // [100000, 512]
  const int*   labels = (const int*)d_in[3];     // [512]
  const int*   epoch  = (const int*)d_in[4];     // scalar
  float* out = (float*)d_out;

  float* ws      = (float*)d_ws;
  float* normA   = ws;                                   // 512*512
  float* invw    = normA + (size_t)BATCH * EMB;          // 100000
  float* pmax    = invw + NCLS;                          // 512*782
  float* psum    = pmax + (size_t)BATCH * NT_TILES;      // 512*782
  float* ztarg   = psum + (size_t)BATCH * NT_TILES;      // 512
  float* rowloss = ztarg + BATCH;                        // 512
  // total workspace ~4.7 MB

  pfc_norm_emb<<<BATCH, 256, 0, stream>>>(emb, normA, epoch);
  pfc_wnorm<<<NCLS / 8, 256, 0, stream>>>(w, invw, epoch);
  dim3 grid(NT_TILES, BATCH / TILE_M);
  const size_t smem_bytes = SMEM_FLTS * sizeof(float);   // 57344 B
  pfc_gemm_softmax<<<grid, 256, smem_bytes, stream>>>(normA, w, invw, labels, pmax, psum, ztarg, epoch, 1);
  pfc_gemm_softmax<<<grid, 256, smem_bytes, stream>>>(emb, fcw, invw, labels, pmax, psum, ztarg, epoch, 0);
  pfc_row_reduce<<<BATCH, 256, 0, stream>>>(pmax, psum, ztarg, rowloss);
  pfc_final<<<1, 256, 0, stream>>>(rowloss, out);
}